// GCNVAE_31507880083798
// MI455X (gfx1250) — compile-verified
//
#include <hip/hip_runtime.h>
#include <hip/hip_bf16.h>

typedef float v2f __attribute__((ext_vector_type(2)));
typedef float v8f __attribute__((ext_vector_type(8)));

#define N_NODES 100000

// ---------------------------------------------------------------------------
// Dense GEMM  C[M,N] = act(A)[M,K] * B[K,N]   (row-major f32)
// One wave computes a 16-row strip across ALL CT=N/16 column tiles.
// Software-pipelined: K-step k+1's A/B fragments are loaded into a separate
// register set while K-step k's CT WMMAs issue back-to-back (distinct
// accumulators -> no RAW chains, loads overlap WMMA latency).
// WMMA lane layouts (ISA 7.12.2):
//   A 16x4 : lanes 0-15 -> row M=l, K={k0,k0+1}; lanes 16-31 -> K={k0+2,k0+3}
//   B 4x16 : mirrored striping, N = lane%16 within a 16-col tile
//   C 16x16: VGPR r -> lanes 0-15: M=r; lanes 16-31: M=r+8
// RELU: applies fmaxf(.,0) to A on load (fuses previous layer's activation).
// ---------------------------------------------------------------------------
template <int CT, bool RELU>
__global__ void __launch_bounds__(128)
gemm_wmma_strip(const float* __restrict__ A, const float* __restrict__ B,
                float* __restrict__ C, int K, int N, int rowTiles) {
    int wave = blockIdx.x * (blockDim.x >> 5) + (threadIdx.x >> 5);
    if (wave >= rowTiles) return;              // wave-uniform: EXEC all-ones below
    int lane = threadIdx.x & 31;
    int half = lane >> 4;                      // 0: lanes 0-15, 1: lanes 16-31
    int l    = lane & 15;
    int row0 = wave << 4;

    const float* ap = A + (size_t)(row0 + l) * K + 2 * half;
    const float* bp = B + (size_t)(2 * half) * N + l;

    v8f acc[CT];
#pragma unroll
    for (int t = 0; t < CT; ++t) acc[t] = v8f{};

    // Prologue: load K-step 0 fragments.
    v2f a_cur;
    v2f b_cur[CT];
    a_cur.x = ap[0];
    a_cur.y = ap[1];
#pragma unroll
    for (int t = 0; t < CT; ++t) {
        b_cur[t].x = bp[16 * t];
        b_cur[t].y = bp[16 * t + N];
    }

    // Steady state: issue next step's loads, then CT WMMAs on current step.
    for (int k0 = 0; k0 < K - 4; k0 += 4) {
        ap += 4;
        bp += (size_t)4 * N;
        v2f a_nxt;
        a_nxt.x = ap[0];
        a_nxt.y = ap[1];
        v2f b_nxt[CT];
#pragma unroll
        for (int t = 0; t < CT; ++t) {
            b_nxt[t].x = bp[16 * t];
            b_nxt[t].y = bp[16 * t + N];
        }

        v2f a = a_cur;
        if (RELU) { a.x = fmaxf(a.x, 0.0f); a.y = fmaxf(a.y, 0.0f); }
#pragma unroll
        for (int t = 0; t < CT; ++t) {
            acc[t] = __builtin_amdgcn_wmma_f32_16x16x4_f32(
                /*neg_a=*/false, a, /*neg_b=*/false, b_cur[t],
                /*c_mod=*/(short)0, acc[t], /*reuse_a=*/false, /*reuse_b=*/false);
        }

        a_cur = a_nxt;
#pragma unroll
        for (int t = 0; t < CT; ++t) b_cur[t] = b_nxt[t];
    }

    // Epilogue: final K-step.
    {
        v2f a = a_cur;
        if (RELU) { a.x = fmaxf(a.x, 0.0f); a.y = fmaxf(a.y, 0.0f); }
#pragma unroll
        for (int t = 0; t < CT; ++t) {
            acc[t] = __builtin_amdgcn_wmma_f32_16x16x4_f32(
                false, a, false, b_cur[t], (short)0, acc[t], false, false);
        }
    }

#pragma unroll
    for (int t = 0; t < CT; ++t) {
        float* cp = C + (size_t)(row0 + half * 8) * N + 16 * t + l;
#pragma unroll
        for (int r = 0; r < 8; ++r) cp[(size_t)r * N] = acc[t][r];
    }
}

// ---------------------------------------------------------------------------
// SpMM edge scatter: out[dst] += w * sup[src]   (per 4-feature chunk)
// grid.x covers edges, grid.y covers F/4 chunks
// ---------------------------------------------------------------------------
__global__ void __launch_bounds__(256)
spmm_scatter(const int* __restrict__ esrc, const int* __restrict__ edst,
             const float* __restrict__ ew, const float* __restrict__ sup,
             float* __restrict__ out, int F, int nE) {
    int e = blockIdx.x * blockDim.x + threadIdx.x;
    if (e >= nE) return;
    int c4 = blockIdx.y << 2;
    int s = esrc[e];
    int d = edst[e];
    float w = ew[e];
    const float4 v = *(const float4*)(sup + (size_t)s * F + c4);  // 16B aligned
    float* op = out + (size_t)d * F + c4;
    atomicAdd(op + 0, w * v.x);
    atomicAdd(op + 1, w * v.y);
    atomicAdd(op + 2, w * v.z);
    atomicAdd(op + 3, w * v.w);
}

// ---------------------------------------------------------------------------
// Elementwise helpers (grid-stride)
// ---------------------------------------------------------------------------
__global__ void zero_f32(float* __restrict__ p, size_t n) {
    size_t i = (size_t)blockIdx.x * blockDim.x + threadIdx.x;
    size_t stride = (size_t)gridDim.x * blockDim.x;
    for (; i < n; i += stride) p[i] = 0.0f;
}

__global__ void copy_f32(float* __restrict__ dst, const float* __restrict__ src, size_t n) {
    size_t i = (size_t)blockIdx.x * blockDim.x + threadIdx.x;
    size_t stride = (size_t)gridDim.x * blockDim.x;
    for (; i < n; i += stride) dst[i] = src[i];
}

// ---------------------------------------------------------------------------
extern "C" void kernel_launch(void* const* d_in, const int* in_sizes, int n_in,
                              void* d_out, int out_size, void* d_ws, size_t ws_size,
                              hipStream_t stream) {
    const float* x    = (const float*)d_in[0];   // [N, 512]
    const int*   esrc = (const int*)  d_in[1];   // [E]
    const int*   edst = (const int*)  d_in[2];   // [E]
    const float* ew   = (const float*)d_in[3];   // [E]
    const float* W1   = (const float*)d_in[4];   // [512,128]
    const float* W2   = (const float*)d_in[5];   // [128, 64]
    const float* W3   = (const float*)d_in[6];   // [ 64, 32]
    const float* Wmu  = (const float*)d_in[7];   // [ 32, 16]
    const float* Wlv  = (const float*)d_in[8];   // [ 32, 16]
    const int nE = in_sizes[1];
    const int N  = N_NODES;
    const int rowTiles = N / 16;                 // 6250
    const int gemmBlocks = (rowTiles + 3) / 4;   // 4 waves per block

    // Workspace: S (support, up to N*128 f32) | H (pre-ReLU activations)
    float* S = (float*)d_ws;
    float* H = (float*)((char*)d_ws + (size_t)N * 128 * sizeof(float));

    float* z  = (float*)d_out;              // segment 0: z (== mu)
    float* mu = z  + (size_t)N * 16;        // segment 1: mu
    float* lv = mu + (size_t)N * 16;        // segment 2: logvar

    auto spmm = [&](const float* sup, float* out, int F) {
        size_t n = (size_t)N * F;
        int zb = (int)((n + 255) / 256); if (zb > 32768) zb = 32768;
        zero_f32<<<zb, 256, 0, stream>>>(out, n);
        dim3 g((nE + 255) / 256, F / 4);
        spmm_scatter<<<g, 256, 0, stream>>>(esrc, edst, ew, sup, out, F, nE);
    };

    // Layer 1: support = x @ W1 ; H = spmm(support)            [N,128]
    gemm_wmma_strip<8, false><<<gemmBlocks, 128, 0, stream>>>(x, W1, S, 512, 128, rowTiles);
    spmm(S, H, 128);

    // Layer 2: support = relu(H) @ W2 ; H = spmm(support)      [N,64]
    gemm_wmma_strip<4, true><<<gemmBlocks, 128, 0, stream>>>(H, W2, S, 128, 64, rowTiles);
    spmm(S, H, 64);

    // Layer 3: support = relu(H) @ W3 ; H = spmm(support)      [N,32]
    gemm_wmma_strip<2, true><<<gemmBlocks, 128, 0, stream>>>(H, W3, S, 64, 32, rowTiles);
    spmm(S, H, 32);

    // mu = spmm(relu(H) @ Wmu)                                 [N,16]
    gemm_wmma_strip<1, true><<<gemmBlocks, 128, 0, stream>>>(H, Wmu, S, 32, 16, rowTiles);
    spmm(S, mu, 16);

    // logvar = spmm(relu(H) @ Wlv)                             [N,16]
    float* Slv = S + (size_t)N * 16;
    gemm_wmma_strip<1, true><<<gemmBlocks, 128, 0, stream>>>(H, Wlv, Slv, 32, 16, rowTiles);
    spmm(Slv, lv, 16);

    // z = mu
    {
        size_t n = (size_t)N * 16;
        int b = (int)((n + 255) / 256); if (b > 32768) b = 32768;
        copy_f32<<<b, 256, 0, stream>>>(z, mu, n);
    }
}